// Conv2d_int8_est_T_87677462380754
// MI455X (gfx1250) — compile-verified
//
#include <hip/hip_runtime.h>
#include <stdint.h>

// Problem constants (from reference)
#define B_     32
#define CIN_   64
#define COUT_  64
#define H_     64
#define W_     64
#define K_     3
#define P_     (CIN_ * K_ * K_)        // 576
#define LUT_STRIDE 257                 // padded dwords per LUT row -> bank = (a+w)%64
#define LDS_BYTES  (256 * LUT_STRIDE * 4)   // 263168 bytes < 320KB WGP LDS

typedef unsigned int uint32x4 __attribute__((ext_vector_type(4)));
typedef int          int32x8  __attribute__((ext_vector_type(8)));
typedef int          int32x4  __attribute__((ext_vector_type(4)));

// ---------------------------------------------------------------------------
// Main: LUT in LDS (loaded by TDM with pad stride 257), per-lane gathers.
// One lane = one output pixel; 64 accumulators (one per cout) in VGPRs.
// (Defined first so the disasm snippet shows the TDM + DS path.)
// ---------------------------------------------------------------------------
__global__ __launch_bounds__(512) void conv_lut_kernel(
    const float*  __restrict__ x,
    const float*  __restrict__ lut,
    const uint8_t* __restrict__ wt,
    const float*  __restrict__ bias,
    float*        __restrict__ out)
{
  extern __shared__ float lds_lut[];   // [256][257] padded LUT

  // --- TDM: DMA the 256x256 f32 LUT into LDS with +1-dword padding per row.
  // Issued by wave 0 only (tensor ops ignore EXEC; one op per wave).
  if (threadIdx.x < 32) {
    uint64_t ga = (uint64_t)(uintptr_t)lut;
    uint32x4 g0;
    g0[0] = 1u;                                   // count=1, user descriptor
    g0[1] = 0u;                                   // lds_addr = 0 (dynamic LDS base)
    g0[2] = (unsigned)(ga & 0xFFFFFFFFu);         // global_addr[31:0]
    g0[3] = (unsigned)((ga >> 32) & 0x1FFFFFFu)   // global_addr[56:32]
          | (2u << 30);                           // type = 2 ("image")
    int32x8 g1;
    g1[0] = (2 << 16)      // data_size = 4 bytes
          | (1 << 20)      // pad_enable
          | (7 << 22);     // pad_interval: every 256 dwords
                           // pad_amount field = 0 -> 1 dword pad  => stride 257
    g1[1] = (256 << 16);   // tensor_dim0[15:0] in bits[63:48]
    g1[2] = (256 << 16);   // tensor_dim1[15:0] in bits[95:80]
    g1[3] = (256 << 16);   // tile_dim0 = 256 in bits[127:112]
    g1[4] = 256;           // tile_dim1 = 256; tile_dim2 = 0
    g1[5] = 256;           // tensor_dim0_stride[31:0] = 256 elements
    g1[6] = 0;
    g1[7] = 0;
    int32x4 z4 = {0, 0, 0, 0};          // groups 2/3 unused (2D tensor)
    int32x8 z8 = {0, 0, 0, 0, 0, 0, 0, 0};
    __builtin_amdgcn_tensor_load_to_lds(g0, g1, z4, z4, z8, 0);
    __builtin_amdgcn_s_wait_tensorcnt(0);
  }
  __syncthreads();

  // --- Pixel mapping: lane -> consecutive x (coalesced loads, spread LDS banks)
  int t  = blockIdx.x * blockDim.x + threadIdx.x;   // 0 .. B*H*W-1
  int xc = t & (W_ - 1);
  int y  = (t >> 6) & (H_ - 1);
  int b  = t >> 12;

  float acc[COUT_];
#pragma unroll
  for (int c = 0; c < COUT_; ++c) acc[c] = 0.0f;

  const float* xb = x + (size_t)b * CIN_ * H_ * W_;

#pragma unroll 1
  for (int cin = 0; cin < CIN_; ++cin) {
#pragma unroll 1
    for (int ky = 0; ky < K_; ++ky) {
      int yy = y + ky - 1;
      bool yok = ((unsigned)yy < (unsigned)H_);
#pragma unroll
      for (int kx = 0; kx < K_; ++kx) {
        int xx = xc + kx - 1;
        bool ok = yok && ((unsigned)xx < (unsigned)W_);
        // Zero-padded patch of *quantized* x: OOB -> q=0 -> index a=128.
        float v = ok ? xb[(cin * H_ + yy) * W_ + xx] : 0.0f;
        float q = fminf(fmaxf(rintf(v * 31.75f), -128.0f), 127.0f); // 127/TF, TF=4
        unsigned a    = (unsigned)((int)q + 128);
        unsigned rowi = a * (unsigned)LUT_STRIDE;   // dword index of LUT row

        int p = cin * 9 + ky * 3 + kx;
        // Uniform address -> scalar loads; 16 dwords = 64 weight bytes for this p.
        const uint32_t* w4 = (const uint32_t*)(wt + p * COUT_);
#pragma unroll
        for (int ci = 0; ci < 16; ++ci) {
          uint32_t wv = w4[ci];
          acc[4 * ci + 0] += lds_lut[rowi + (wv         & 0xffu)];
          acc[4 * ci + 1] += lds_lut[rowi + ((wv >> 8)  & 0xffu)];
          acc[4 * ci + 2] += lds_lut[rowi + ((wv >> 16) & 0xffu)];
          acc[4 * ci + 3] += lds_lut[rowi + ( wv >> 24        )];
        }
      }
    }
  }

  // --- Epilogue: dequantize scale + bias, strided stores (coalesced over x).
  const float SCALE = (4.0f * 1.0f) / (127.0f * 127.0f);
  float* ob = out + (size_t)b * COUT_ * H_ * W_ + (size_t)y * W_ + xc;
#pragma unroll
  for (int c = 0; c < COUT_; ++c)
    ob[(size_t)c * H_ * W_] = acc[c] * SCALE + bias[c];
}

// ---------------------------------------------------------------------------
// Prep: quantize weights and transpose to [p][cout] u8 (value = q+128).
// p = cin*9 + ky*3 + kx, matching conv_general_dilated_patches channel-major.
// ---------------------------------------------------------------------------
__global__ void quant_weights_kernel(const float* __restrict__ w,
                                     uint8_t* __restrict__ wt) {
  int c = blockIdx.x;    // 0..63
  int p = threadIdx.x;   // 0..575
  if (p < P_) {
    float v = w[c * P_ + p];
    float q = fminf(fmaxf(rintf(v * 127.0f), -128.0f), 127.0f); // TW = 1.0
    wt[p * COUT_ + c] = (uint8_t)((int)q + 128);
  }
}

// ---------------------------------------------------------------------------
extern "C" void kernel_launch(void* const* d_in, const int* in_sizes, int n_in,
                              void* d_out, int out_size, void* d_ws, size_t ws_size,
                              hipStream_t stream) {
  (void)in_sizes; (void)n_in; (void)out_size; (void)ws_size;
  const float* x    = (const float*)d_in[0];   // [32,64,64,64]
  const float* w    = (const float*)d_in[1];   // [64,64,3,3]
  const float* lut  = (const float*)d_in[2];   // [256,256]
  /* d_in[3] = gradient_lut (unused in forward) */
  const float* bias = (const float*)d_in[4];   // [64]
  float* out = (float*)d_out;                  // [32,64,64,64]

  uint8_t* wt = (uint8_t*)d_ws;                // [576][64] u8, 36864 bytes

  quant_weights_kernel<<<COUT_, P_, 0, stream>>>(w, wt);

  dim3 grid((B_ * H_ * W_) / 512);
  conv_lut_kernel<<<grid, 512, LDS_BYTES, stream>>>(x, lut, wt, bias, out);
}